// FM_MTL_48043504173401
// MI455X (gfx1250) — compile-verified
//
#include <hip/hip_runtime.h>
#include <hip/hip_bf16.h>

// DeepFM inference: first-order (dense linear + 1-d sparse embeddings) +
// FM second-order 0.5*(||sum_f v||^2 - sum_f ||v||^2), two sigmoid heads.
//
// Layout: 16 lanes per batch row (E==16), 2 rows per wave32, 16 rows per wave
// tile. fm_emb gathers are perfectly coalesced 64B requests. Per-wave the
// summed vectors form a 16x16 f32 tile S in LDS; ||s||^2 per row is computed
// as diag(S * S^T) via 4x V_WMMA_F32_16X16X4_F32 (exact f32 WMMA).

typedef __attribute__((ext_vector_type(2))) float v2f;
typedef __attribute__((ext_vector_type(8))) float v8f;

namespace {
constexpr int BT   = 16384;
constexpr int FSP  = 26;
constexpr int FDN  = 13;
constexpr int VOC  = 100000;
constexpr int EMB  = 16;
constexpr int ROWS_PER_BLOCK = 128;   // 8 waves * 16 rows
}

__global__ __launch_bounds__(256) void fm_mtl_kernel(
    const int*   __restrict__ sp,    // [B, 26] sparse indices
    const float* __restrict__ dn,    // [B, 13] dense inputs
    const float* __restrict__ Wd,    // [13, 1] dense linear W
    const float* __restrict__ bd,    // [1]     dense linear b
    const float* __restrict__ lem,   // [26, 100000, 1] linear embeddings
    const float* __restrict__ fme,   // [26, 100000, 16] FM embeddings
    const float* __restrict__ fW, const float* __restrict__ fb,
    const float* __restrict__ lW, const float* __restrict__ lb,
    float*       __restrict__ out)   // [B, 2]
{
    // Per-wave tiles. S row stride 18 floats: 8B aligned and bank-conflict
    // free for the strided v2f reads in the WMMA setup (gcd trick mod 64).
    __shared__ float Sh[8][16][18];
    __shared__ float Fo[8][16];   // first-order total per row (no bias yet)
    __shared__ float Qo[8][16];   // sum_f sum_e v^2 per row
    __shared__ float Dg[8][16];   // ||sum_f v||^2 per row (WMMA diag)

    const int tid = threadIdx.x;
    const int w   = tid >> 5;          // wave in block (0..7)
    const int l   = tid & 31;          // lane in wave
    const int e   = l & 15;            // embedding element handled by lane
    const int h   = l >> 4;            // half-wave selector
    const int waveRow = blockIdx.x * ROWS_PER_BLOCK + w * 16;

    const float wdn = (e < FDN) ? Wd[e] : 0.0f;

    // ---- Phase 1: gather + accumulate. 2 rows per wave per iteration. ----
    for (int rp = 0; rp < 8; ++rp) {
        const int r  = rp * 2 + h;             // row in wave tile (0..15)
        const int gr = waveRow + r;            // global batch row
        const int* row_idx = sp + (size_t)gr * FSP;

        float s = 0.0f, q = 0.0f;
        #pragma unroll
        for (int f = 0; f < FSP; ++f) {
            const int idx = row_idx[f];        // uniform across the 16 lanes
            // 16 lanes read 16 consecutive floats -> one 64B request
            const float v = fme[((size_t)f * VOC + (size_t)idx) * EMB + e];
            s += v;
            q = fmaf(v, v, q);
        }

        // first-order: distribute the 26 1-d gathers + 13 dense terms
        float lin = 0.0f;
        #pragma unroll
        for (int t = 0; t < 2; ++t) {
            const int f2 = e + 16 * t;
            if (f2 < FSP) {
                const int idx2 = row_idx[f2];
                lin += lem[(size_t)f2 * VOC + (size_t)idx2];
            }
        }
        if (e < FDN) lin = fmaf(dn[(size_t)gr * FDN + e], wdn, lin);

        // butterfly reduce q and lin over the 16-lane half-wave
        #pragma unroll
        for (int m = 8; m >= 1; m >>= 1) {
            q   += __shfl_xor(q,   m, 16);
            lin += __shfl_xor(lin, m, 16);
        }

        Sh[w][r][e] = s;
        if (e == 0) { Fo[w][r] = lin; Qo[w][r] = q; }
    }
    __syncthreads();

    // ---- Phase 2: diag(S * S^T) via V_WMMA_F32_16X16X4_F32 (K looped 4x).
    // A(16x4) layout: VGPR0 = {K0 | K2}, VGPR1 = {K1 | K3}, lanes = M.
    // B(4x16) = S^T has an identical register image, so SRC0 == SRC1.
    v8f c = {0.f, 0.f, 0.f, 0.f, 0.f, 0.f, 0.f, 0.f};
    {
        const int m = e; // A-matrix row for this lane
        #pragma unroll
        for (int c4 = 0; c4 < 4; ++c4) {
            const v2f a = *(const v2f*)&Sh[w][m][c4 * 4 + 2 * h];
            c = __builtin_amdgcn_wmma_f32_16x16x4_f32(
                    /*neg_a=*/false, a, /*neg_b=*/false, a,
                    /*c_mod=*/(short)0, c, /*reuse_a=*/false, /*reuse_b=*/false);
        }
    }
    // C layout: lanes 0-7 hold D[m][m] in c[m] (m<8); lanes 24-31 hold
    // D[m][m] for m=8..15 in c[m-8].
    {
        float d = 0.0f;
        #pragma unroll
        for (int i = 0; i < 8; ++i) {
            if (l == i)      d = c[i];
            if (l == i + 24) d = c[i];
        }
        if (l < 8)        Dg[w][l] = d;
        else if (l >= 24) Dg[w][l - 16] = d;
    }
    __syncthreads();

    // ---- Phase 3: finalize 16 rows per wave ----
    if (l < 16) {
        const int gr = waveRow + l;
        const float logit = Fo[w][l] + bd[0] + 0.5f * (Dg[w][l] - Qo[w][l]);
        const float zf = fmaf(logit, fW[0], fb[0]);
        const float zl = fmaf(logit, lW[0], lb[0]);
        out[2 * gr + 0] = 1.0f / (1.0f + __expf(-zf));
        out[2 * gr + 1] = 1.0f / (1.0f + __expf(-zl));
    }
}

extern "C" void kernel_launch(void* const* d_in, const int* in_sizes, int n_in,
                              void* d_out, int out_size, void* d_ws, size_t ws_size,
                              hipStream_t stream) {
    (void)in_sizes; (void)n_in; (void)d_ws; (void)ws_size; (void)out_size;
    const int*   sp  = (const int*)  d_in[0];
    const float* dn  = (const float*)d_in[1];
    const float* Wd  = (const float*)d_in[2];
    const float* bd  = (const float*)d_in[3];
    const float* lem = (const float*)d_in[4];
    const float* fme = (const float*)d_in[5];
    const float* fW  = (const float*)d_in[6];
    const float* fb  = (const float*)d_in[7];
    const float* lW  = (const float*)d_in[8];
    const float* lb  = (const float*)d_in[9];
    float* out = (float*)d_out;

    dim3 grid(BT / ROWS_PER_BLOCK);   // 128 blocks
    dim3 block(256);                  // 8 wave32 per block
    fm_mtl_kernel<<<grid, block, 0, stream>>>(sp, dn, Wd, bd, lem, fme,
                                              fW, fb, lW, lb, out);
}